// VanillaRNN_91190745628836
// MI455X (gfx1250) — compile-verified
//
#include <hip/hip_runtime.h>
#include <stdint.h>

typedef float v2f __attribute__((ext_vector_type(2)));
typedef float v8f __attribute__((ext_vector_type(8)));

#define B_ 64
#define T_ 512
#define I_ 1024
#define H_ 1024

// ---------- CDNA5 async Global->LDS staging (ASYNCcnt-tracked) ----------
__device__ __forceinline__ void async_lds_load_b128(uint32_t lds_byte_off, const void* gsrc) {
    uint64_t ga = (uint64_t)(uintptr_t)gsrc;
    asm volatile("global_load_async_to_lds_b128 %0, %1, off"
                 :: "v"(lds_byte_off), "v"(ga) : "memory");
}
__device__ __forceinline__ void wait_async_zero() {
    asm volatile("s_wait_asynccnt 0x0" ::: "memory");
}

// ---------- grid-wide barrier (agent scope, monotonic target) ----------
__device__ __forceinline__ void grid_barrier(unsigned* cnt, unsigned target) {
    __syncthreads();
    if (threadIdx.x == 0) {
        __hip_atomic_fetch_add(cnt, 1u, __ATOMIC_ACQ_REL, __HIP_MEMORY_SCOPE_AGENT);
        while (__hip_atomic_load(cnt, __ATOMIC_ACQUIRE, __HIP_MEMORY_SCOPE_AGENT) < target) {
            __builtin_amdgcn_s_sleep(2);
        }
    }
    __syncthreads();
}

__global__ void rnn_init(unsigned* cnt) {
    if (threadIdx.x == 0) *cnt = 0u;
}

// ---------- Phase 1: out[b*T+t][h] = x[b*T+t][:] . W_ih[h][:] + b_ih[h] ----------
// grid: 16384 blocks x 256 threads. block = (m_group, n_tile); wave w -> M-tile m_group*8+w.
__global__ __launch_bounds__(256) void rnn_xproj(const float* __restrict__ x,
                                                 const float* __restrict__ Wih,
                                                 const float* __restrict__ bih,
                                                 float* __restrict__ out) {
    __shared__ float lds_w[16 * I_];   // 64 KB: W_ih rows n0..n0+15
    const int tid  = threadIdx.x;
    const int lane = tid & 31;
    const int wid  = tid >> 5;
    const int n_tile  = blockIdx.x & 63;
    const int m_group = blockIdx.x >> 6;
    const int n0 = n_tile * 16;
    const int m0 = (m_group * 8 + wid) * 16;

    // async-stage the 16x1024 W slab (contiguous rows) into LDS
    const float* slab = Wih + (size_t)n0 * I_;
    for (int idx = tid; idx < 4096; idx += 256) {
        uint32_t loff = (uint32_t)(uintptr_t)(&lds_w[idx * 4]);
        async_lds_load_b128(loff, slab + idx * 4);
    }
    wait_async_zero();
    __syncthreads();

    // A operand: 16x4 fp32 tile; lane m=lane&15, K-pair base = (lane>>4)*2
    const int kb = (lane >> 4) * 2;
    const float* Ap = x + (size_t)(m0 + (lane & 15)) * I_ + kb;
    const float* Bp = lds_w + (lane & 15) * I_ + kb;

    v8f acc = {};
    for (int k0 = 0; k0 < I_; k0 += 64) {
        __builtin_prefetch(Ap + k0 + 256, 0, 3);   // one near-cache prefetch per 16 WMMAs
#pragma unroll
        for (int k = k0; k < k0 + 64; k += 4) {
            v2f a = *(const v2f*)(Ap + k);
            v2f b = *(const v2f*)(Bp + k);
            acc = __builtin_amdgcn_wmma_f32_16x16x4_f32(false, a, false, b,
                                                        (short)0, acc, false, false);
        }
    }

    const int ncol = n0 + (lane & 15);
    const int mrow = m0 + ((lane >> 4) * 8);
    const float bias = bih[ncol];
#pragma unroll
    for (int r = 0; r < 8; ++r)
        out[(size_t)(mrow + r) * H_ + ncol] = acc[r] + bias;
}

// ---------- Phase 2: persistent recurrence ----------
// grid: 64 blocks (one 16-col slab of W_hh each, LDS-resident) x 128 threads (4 waves = 4 M-tiles of B).
__global__ __launch_bounds__(128) void rnn_recur(const float* __restrict__ h0,
                                                 const float* __restrict__ Whh,
                                                 const float* __restrict__ bhh,
                                                 float* __restrict__ out,
                                                 float* __restrict__ hbuf0,
                                                 float* __restrict__ hbuf1,
                                                 unsigned* __restrict__ cnt) {
    __shared__ float lds_w[16 * H_];   // 64 KB: W_hh rows n0..n0+15, reused for all 512 steps
    const int tid  = threadIdx.x;
    const int lane = tid & 31;
    const int wid  = tid >> 5;
    const int n0 = blockIdx.x * 16;
    const int m0 = wid * 16;

    const float* slab = Whh + (size_t)n0 * H_;
    for (int idx = tid; idx < 4096; idx += 128) {
        uint32_t loff = (uint32_t)(uintptr_t)(&lds_w[idx * 4]);
        async_lds_load_b128(loff, slab + idx * 4);
    }
    // stage this block's h0 column slice into the ping buffer
    for (int i = tid; i < 64 * 16; i += 128) {
        int row = i >> 4;
        int col = n0 + (i & 15);
        hbuf0[row * H_ + col] = h0[row * H_ + col];
    }
    wait_async_zero();
    __threadfence();
    grid_barrier(cnt, 64u);   // all blocks staged h0

    const int kb   = (lane >> 4) * 2;
    const int arow = m0 + (lane & 15);
    const int ncol = n0 + (lane & 15);
    const int mrow = m0 + ((lane >> 4) * 8);
    const float bias = bhh[ncol];
    const float* Bp = lds_w + (lane & 15) * H_ + kb;

    for (int t = 0; t < T_; ++t) {
        const float* cur = (t & 1) ? hbuf1 : hbuf0;
        float*       nxt = (t & 1) ? hbuf0 : hbuf1;
        const float* Ap  = cur + (size_t)arow * H_ + kb;

        v8f acc = {};
        for (int k0 = 0; k0 < H_; k0 += 64) {
            __builtin_prefetch(Ap + k0 + 256, 0, 3);
#pragma unroll
            for (int k = k0; k < k0 + 64; k += 4) {
                v2f a = *(const v2f*)(Ap + k);
                v2f b = *(const v2f*)(Bp + k);
                acc = __builtin_amdgcn_wmma_f32_16x16x4_f32(false, a, false, b,
                                                            (short)0, acc, false, false);
            }
        }

#pragma unroll
        for (int r = 0; r < 8; ++r) {
            size_t oidx = (size_t)(mrow + r) * (size_t)(T_ * H_) + (size_t)t * H_ + ncol;
            float hv = tanhf(out[oidx] + acc[r] + bias);
            out[oidx] = hv;
            nxt[(mrow + r) * H_ + ncol] = hv;
        }
        __threadfence();
        grid_barrier(cnt, 64u * (unsigned)(t + 2));
    }
}

extern "C" void kernel_launch(void* const* d_in, const int* in_sizes, int n_in,
                              void* d_out, int out_size, void* d_ws, size_t ws_size,
                              hipStream_t stream) {
    const float* x   = (const float*)d_in[0];
    const float* h0  = (const float*)d_in[1];
    const float* Wih = (const float*)d_in[2];
    const float* bih = (const float*)d_in[3];
    const float* Whh = (const float*)d_in[4];
    const float* bhh = (const float*)d_in[5];
    float* out = (float*)d_out;

    unsigned* cnt = (unsigned*)d_ws;
    float* hbuf0 = (float*)((char*)d_ws + 256);
    float* hbuf1 = hbuf0 + (size_t)B_ * H_;

    rnn_init<<<dim3(1), dim3(32), 0, stream>>>(cnt);
    rnn_xproj<<<dim3(16384), dim3(256), 0, stream>>>(x, Wih, bih, out);
    rnn_recur<<<dim3(64), dim3(128), 0, stream>>>(h0, Whh, bhh, out, hbuf0, hbuf1, cnt);
}